// MultiModalBindingModel_40604620817088
// MI455X (gfx1250) — compile-verified
//
#include <hip/hip_runtime.h>
#include <hip/hip_bf16.h>

typedef __attribute__((ext_vector_type(2))) float v2f;
typedef __attribute__((ext_vector_type(8))) float v8f;

#define NN     100000
#define EE     800000
#define BB     4096
#define SEQL   1024
#define NODEF  64
#define HH     128
#define LLAY   3

// ---------------------------------------------------------------------------
// Fused GEMM:  Y = [LN](relu(A1@W1 + b1 [+ A2@W2 + b2]))
//   A1: [M,K1] (optionally row-scaled by scale1), W1: [K1,Hout] row-major
//   A2: [M,K2] (optionally row-scaled by scale2), W2: [K2,Hout] row-major
//   LN: per-row layernorm over Hout==128 (requires gridDim.y==1)
// Block: 256 threads = 8 wave32; each wave -> one 16x16 tile via
// v_wmma_f32_16x16x4_f32; block covers 16 rows x 128 cols.
// ---------------------------------------------------------------------------
template<bool DUAL, bool LN>
__global__ __launch_bounds__(256)
void gemm_fused(const float* __restrict__ A1, const float* __restrict__ W1,
                const float* __restrict__ bias1, const float* __restrict__ scale1,
                const float* __restrict__ A2, const float* __restrict__ W2,
                const float* __restrict__ bias2, const float* __restrict__ scale2,
                const float* __restrict__ gamma, const float* __restrict__ beta,
                float* __restrict__ Y, int M, int K1, int K2, int Hout)
{
    __shared__ float As[16][65];                       // A chunk, pad to kill bank conflicts
    __shared__ float Ys[LN ? 16 : 1][LN ? 130 : 2];    // row-resident tile for LN
    __shared__ float s_mean[16], s_rstd[16];

    const int tid  = threadIdx.x;
    const int wave = tid >> 5;
    const int lane = tid & 31;
    const int hi   = lane >> 4;       // half-wave select
    const int l16  = lane & 15;
    const int row0 = blockIdx.x * 16;
    const int colW = blockIdx.y * 128 + wave * 16;     // wave's column base
    const int colH = colW + l16;                       // this thread's column

    // A-stage mapping: each thread loads one float4 of the 16x64 chunk
    const int lr = tid >> 4;          // 0..15  (row in chunk)
    const int lc = (tid & 15) << 2;   // 0..60  (col in chunk)

    v8f c = {};

    // ---- input 1 -----------------------------------------------------------
    {
        const float s = (scale1 != nullptr) ? scale1[row0 + lr] : 1.0f;
        for (int kc = 0; kc < K1; kc += 64) {
            __syncthreads();
            const float4 av = *(const float4*)(A1 + (size_t)(row0 + lr) * K1 + kc + lc);
            As[lr][lc + 0] = av.x * s;  As[lr][lc + 1] = av.y * s;
            As[lr][lc + 2] = av.z * s;  As[lr][lc + 3] = av.w * s;
            __syncthreads();
            #pragma unroll
            for (int kk = 0; kk < 64; kk += 4) {
                v2f a, b;
                a.x = As[l16][kk + 2*hi + 0];          // ISA f32 A-fragment layout
                a.y = As[l16][kk + 2*hi + 1];
                const float* wp = W1 + (size_t)(kc + kk + 2*hi) * Hout + colH;
                b.x = wp[0];
                b.y = wp[Hout];
                c = __builtin_amdgcn_wmma_f32_16x16x4_f32(
                        false, a, false, b, (short)0, c, false, false);
            }
        }
    }
    // ---- input 2 (graph layer / fuse concat half) --------------------------
    if (DUAL) {
        const float s = (scale2 != nullptr) ? scale2[row0 + lr] : 1.0f;
        for (int kc = 0; kc < K2; kc += 64) {
            __syncthreads();
            const float4 av = *(const float4*)(A2 + (size_t)(row0 + lr) * K2 + kc + lc);
            As[lr][lc + 0] = av.x * s;  As[lr][lc + 1] = av.y * s;
            As[lr][lc + 2] = av.z * s;  As[lr][lc + 3] = av.w * s;
            __syncthreads();
            #pragma unroll
            for (int kk = 0; kk < 64; kk += 4) {
                v2f a, b;
                a.x = As[l16][kk + 2*hi + 0];
                a.y = As[l16][kk + 2*hi + 1];
                const float* wp = W2 + (size_t)(kc + kk + 2*hi) * Hout + colH;
                b.x = wp[0];
                b.y = wp[Hout];
                c = __builtin_amdgcn_wmma_f32_16x16x4_f32(
                        false, a, false, b, (short)0, c, false, false);
            }
        }
    }

    // ---- epilogue: bias + relu (+ layernorm) -------------------------------
    float bv = 0.f;
    if (bias1) bv += bias1[colH];
    if (DUAL && bias2 != nullptr) bv += bias2[colH];

    if (LN) {
        #pragma unroll
        for (int r = 0; r < 8; ++r) {
            float v = fmaxf(c[r] + bv, 0.f);
            Ys[r + hi * 8][wave * 16 + l16] = v;       // C/D layout: VGPR r -> row r / r+8
        }
        __syncthreads();
        if (tid < 16) {
            float s = 0.f, s2 = 0.f;
            for (int j = 0; j < 128; ++j) { float x = Ys[tid][j]; s += x; s2 += x * x; }
            float m   = s * (1.f / 128.f);
            float var = s2 * (1.f / 128.f) - m * m;
            s_mean[tid] = m;
            s_rstd[tid] = rsqrtf(var + 1e-5f);
        }
        __syncthreads();
        #pragma unroll
        for (int e = 0; e < 8; ++e) {
            int idx = e * 256 + tid;
            int rr = idx >> 7, cc = idx & 127;
            float x = (Ys[rr][cc] - s_mean[rr]) * s_rstd[rr];
            Y[(size_t)(row0 + rr) * 128 + cc] = x * gamma[cc] + beta[cc];
        }
    } else {
        #pragma unroll
        for (int r = 0; r < 8; ++r) {
            float v = fmaxf(c[r] + bv, 0.f);
            Y[(size_t)(row0 + r + hi * 8) * Hout + colH] = v;
        }
    }
}

// ---------------------------------------------------------------------------
// CSR construction (once per tower; edges are layer-invariant)
// ---------------------------------------------------------------------------
__global__ void count_int_kernel(const int* __restrict__ idx, int* __restrict__ cnt, int n) {
    int i = blockIdx.x * blockDim.x + threadIdx.x;
    if (i < n) atomicAdd(&cnt[idx[i]], 1);
}

// single-block exclusive scan (Hillis-Steele over 1024-wide chunks, shared carry)
__global__ __launch_bounds__(1024)
void scan_kernel(const int* __restrict__ cnt, int* __restrict__ rowptr, int n) {
    __shared__ int sh[1024];
    __shared__ int carry;
    int tid = threadIdx.x;
    if (tid == 0) carry = 0;
    __syncthreads();
    for (int base = 0; base < n; base += 1024) {
        int i = base + tid;
        int v = (i < n) ? cnt[i] : 0;
        __syncthreads();                 // protect sh reuse across iterations
        sh[tid] = v;
        __syncthreads();
        #pragma unroll
        for (int off = 1; off < 1024; off <<= 1) {
            int t = (tid >= off) ? sh[tid - off] : 0;
            __syncthreads();
            sh[tid] += t;
            __syncthreads();
        }
        int c = carry;
        if (i < n) rowptr[i] = c + sh[tid] - v;   // exclusive
        __syncthreads();
        if (tid == 0) carry = c + sh[1023];
        __syncthreads();
    }
    if (tid == 0) rowptr[n] = carry;
}

__global__ void fill_csr_kernel(const int* __restrict__ edge, const int* __restrict__ rowptr,
                                int* __restrict__ fill, int* __restrict__ csr, int ne) {
    int e = blockIdx.x * blockDim.x + threadIdx.x;
    if (e >= ne) return;
    int dst = edge[ne + e];
    int pos = atomicAdd(&fill[dst], 1);
    csr[rowptr[dst] + pos] = edge[e];   // store src node id, grouped by dst
}

// ---------------------------------------------------------------------------
// Mean neighbor aggregation as a pure gather: one wave32 per node, each lane
// owns 4 channels (float4). h is L2-resident (51MB << 192MB L2). No atomics.
// ---------------------------------------------------------------------------
__global__ __launch_bounds__(256)
void agg_gather_kernel(const float* __restrict__ h, const int* __restrict__ rowptr,
                       const int* __restrict__ csr, float* __restrict__ agg, int n) {
    int node = blockIdx.x * 8 + (threadIdx.x >> 5);
    if (node >= n) return;
    int c  = (threadIdx.x & 31) << 2;
    int s0 = rowptr[node], s1 = rowptr[node + 1];
    float4 acc = make_float4(0.f, 0.f, 0.f, 0.f);
    for (int j = s0; j < s1; ++j) {
        int src = csr[j];
        const float4 v = *(const float4*)(h + (size_t)src * HH + c);
        acc.x += v.x; acc.y += v.y; acc.z += v.z; acc.w += v.w;
    }
    float inv = 1.0f / fmaxf((float)(s1 - s0), 1.0f);   // mean with deg clip(1)
    acc.x *= inv; acc.y *= inv; acc.z *= inv; acc.w *= inv;
    *(float4*)(agg + (size_t)node * HH + c) = acc;
}

// batch is sorted -> graph b owns the contiguous node range [rowptr[b], rowptr[b+1])
__global__ __launch_bounds__(256)
void pool_gather_kernel(const float* __restrict__ h, const int* __restrict__ rowptr,
                        float* __restrict__ pooled, int nb) {
    int b = blockIdx.x * 8 + (threadIdx.x >> 5);
    if (b >= nb) return;
    int c  = (threadIdx.x & 31) << 2;
    int s0 = rowptr[b], s1 = rowptr[b + 1];
    float4 acc = make_float4(0.f, 0.f, 0.f, 0.f);
    for (int j = s0; j < s1; ++j) {
        const float4 v = *(const float4*)(h + (size_t)j * HH + c);
        acc.x += v.x; acc.y += v.y; acc.z += v.z; acc.w += v.w;
    }
    float inv = 1.0f / fmaxf((float)(s1 - s0), 1.0f);   // mean with count clip(1)
    acc.x *= inv; acc.y *= inv; acc.z *= inv; acc.w *= inv;
    *(float4*)(pooled + (size_t)b * HH + c) = acc;
}

// ---------------------------------------------------------------------------
__global__ void feat_kernel(const float* __restrict__ tcr, const float* __restrict__ pep,
                            float* __restrict__ feat) {
    int i = blockIdx.x * blockDim.x + threadIdx.x;
    if (i >= BB * HH) return;
    int b = i >> 7, c = i & 127;
    float a = tcr[i], p = pep[i];
    float* f = feat + (size_t)b * (4 * HH);
    f[c]          = a;
    f[HH + c]     = p;
    f[2 * HH + c] = fabsf(a - p);
    f[3 * HH + c] = a * p;
}

__global__ void logits_kernel(const float* __restrict__ hid, const float* __restrict__ W2,
                              const float* __restrict__ b2, float* __restrict__ out) {
    int i = blockIdx.x * blockDim.x + threadIdx.x;
    if (i >= BB * 2) return;
    int b = i >> 1, j = i & 1;
    const float* hp = hid + (size_t)b * 256;
    float s = b2[j];
    for (int k = 0; k < 256; ++k) s += hp[k] * W2[k * 2 + j];
    out[i] = s;
}

// ---------------------------------------------------------------------------
extern "C" void kernel_launch(void* const* d_in, const int* in_sizes, int n_in,
                              void* d_out, int out_size, void* d_ws, size_t ws_size,
                              hipStream_t stream) {
    const float* tcr_seq  = (const float*)d_in[0];
    const float* pep_seq  = (const float*)d_in[1];
    const float* tcr_x    = (const float*)d_in[2];
    const float* pep_x    = (const float*)d_in[3];
    const int*   tcr_edge = (const int*)d_in[4];
    const int*   pep_edge = (const int*)d_in[5];
    const int*   tcr_bat  = (const int*)d_in[6];
    const int*   pep_bat  = (const int*)d_in[7];
    const float* seq_W    = (const float*)d_in[8];
    const float* seq_b    = (const float*)d_in[9];
    const float* seq_g    = (const float*)d_in[10];
    const float* seq_be   = (const float*)d_in[11];
    const float* gin_W    = (const float*)d_in[12];
    const float* gin_b    = (const float*)d_in[13];
    const float* gin_g    = (const float*)d_in[14];
    const float* gin_be   = (const float*)d_in[15];
    const float* glsW     = (const float*)d_in[16];
    const float* glsb     = (const float*)d_in[17];
    const float* glnW     = (const float*)d_in[18];
    const float* glnb     = (const float*)d_in[19];
    const float* glg      = (const float*)d_in[20];
    const float* glbe     = (const float*)d_in[21];
    const float* gout_W   = (const float*)d_in[22];
    const float* gout_b   = (const float*)d_in[23];
    const float* gout_g   = (const float*)d_in[24];
    const float* gout_be  = (const float*)d_in[25];
    const float* fuse_W   = (const float*)d_in[26];
    const float* fuse_b   = (const float*)d_in[27];
    const float* fuse_g   = (const float*)d_in[28];
    const float* fuse_be  = (const float*)d_in[29];
    const float* bh_W1    = (const float*)d_in[30];
    const float* bh_b1    = (const float*)d_in[31];
    const float* bh_W2    = (const float*)d_in[32];
    const float* bh_b2    = (const float*)d_in[33];

    // workspace layout
    const size_t NH = (size_t)NN * HH;
    const size_t BH = (size_t)BB * HH;
    float* ws     = (float*)d_ws;
    float* hbuf   = ws;
    float* agg    = hbuf + NH;
    float* hnew   = agg + NH;
    float* pooled = hnew + NH;                 // BH
    float* feat   = pooled + BH;               // BB*512
    float* hid    = feat + (size_t)BB * 512;   // BB*256
    int*   cnt    = (int*)(hid + (size_t)BB * 256);  // NN
    int*   rowptr = cnt + NN;                  // NN+1
    int*   fillc  = rowptr + NN + 1;           // NN
    int*   csr    = fillc + NN;                // EE

    // output layout: logits, z_tcr_seq, z_tcr_g, z_pep_seq, z_pep_g, tcr, pep
    float* out       = (float*)d_out;
    float* logits    = out;
    float* z_seq[2]  = { out + 8192,          out + 8192 + 2 * BH };
    float* z_g[2]    = { out + 8192 + BH,     out + 8192 + 3 * BH };
    float* fuse_o[2] = { out + 8192 + 4 * BH, out + 8192 + 5 * BH };

    dim3 blk(256);

    // ---- sequence towers ---------------------------------------------------
    for (int i = 0; i < 2; ++i) {
        gemm_fused<false, true><<<dim3(BB / 16, 1), blk, 0, stream>>>(
            i ? pep_seq : tcr_seq, seq_W + (size_t)i * SEQL * HH, seq_b + i * HH, nullptr,
            nullptr, nullptr, nullptr, nullptr,
            seq_g + i * HH, seq_be + i * HH, z_seq[i], BB, SEQL, 0, HH);
    }

    // ---- graph towers ------------------------------------------------------
    for (int i = 0; i < 2; ++i) {
        const float* x     = i ? pep_x    : tcr_x;
        const int*   edge  = i ? pep_edge : tcr_edge;
        const int*   batch = i ? pep_bat  : tcr_bat;

        gemm_fused<false, true><<<dim3(NN / 16, 1), blk, 0, stream>>>(
            x, gin_W + (size_t)i * NODEF * HH, gin_b + i * HH, nullptr,
            nullptr, nullptr, nullptr, nullptr,
            gin_g + i * HH, gin_be + i * HH, hbuf, NN, NODEF, 0, HH);

        // build CSR by destination (once; reused by all 3 layers)
        hipMemsetAsync(cnt, 0, NN * sizeof(int), stream);
        count_int_kernel<<<(EE + 255) / 256, blk, 0, stream>>>(edge + EE, cnt, EE);
        scan_kernel<<<1, 1024, 0, stream>>>(cnt, rowptr, NN);
        hipMemsetAsync(fillc, 0, NN * sizeof(int), stream);
        fill_csr_kernel<<<(EE + 255) / 256, blk, 0, stream>>>(edge, rowptr, fillc, csr, EE);

        float* hc = hbuf;
        float* hn = hnew;
        for (int l = 0; l < LLAY; ++l) {
            agg_gather_kernel<<<(NN + 7) / 8, blk, 0, stream>>>(hc, rowptr, csr, agg, NN);
            size_t off = (size_t)i * LLAY + l;
            gemm_fused<true, true><<<dim3(NN / 16, 1), blk, 0, stream>>>(
                hc,  glsW + off * HH * HH, glsb + off * HH, nullptr,
                agg, glnW + off * HH * HH, glnb + off * HH, nullptr,
                glg + off * HH, glbe + off * HH, hn, NN, HH, HH, HH);
            float* t = hc; hc = hn; hn = t;
        }

        // mean pooling: batch is sorted -> contiguous per-graph node ranges
        hipMemsetAsync(cnt, 0, BB * sizeof(int), stream);
        count_int_kernel<<<(NN + 255) / 256, blk, 0, stream>>>(batch, cnt, NN);
        scan_kernel<<<1, 1024, 0, stream>>>(cnt, rowptr, BB);
        pool_gather_kernel<<<(BB + 7) / 8, blk, 0, stream>>>(hc, rowptr, pooled, BB);

        gemm_fused<false, true><<<dim3(BB / 16, 1), blk, 0, stream>>>(
            pooled, gout_W + (size_t)i * HH * HH, gout_b + i * HH, nullptr,
            nullptr, nullptr, nullptr, nullptr,
            gout_g + i * HH, gout_be + i * HH, z_g[i], BB, HH, 0, HH);
    }

    // ---- fuse (concat GEMM as dual accumulation) ---------------------------
    for (int i = 0; i < 2; ++i) {
        gemm_fused<true, true><<<dim3(BB / 16, 1), blk, 0, stream>>>(
            z_seq[i], fuse_W + (size_t)i * 2 * HH * HH,           fuse_b + i * HH, nullptr,
            z_g[i],   fuse_W + (size_t)i * 2 * HH * HH + HH * HH, nullptr,         nullptr,
            fuse_g + i * HH, fuse_be + i * HH, fuse_o[i], BB, HH, HH, HH);
    }

    // ---- binding head ------------------------------------------------------
    feat_kernel<<<(BB * HH + 255) / 256, blk, 0, stream>>>(fuse_o[0], fuse_o[1], feat);
    gemm_fused<false, false><<<dim3(BB / 16, 2), blk, 0, stream>>>(
        feat, bh_W1, bh_b1, nullptr,
        nullptr, nullptr, nullptr, nullptr,
        nullptr, nullptr, hid, BB, 4 * HH, 0, 256);
    logits_kernel<<<(BB * 2 + 255) / 256, blk, 0, stream>>>(hid, bh_W2, bh_b2, logits);
}